// MSBEGCL_Encoder_566935683766
// MI455X (gfx1250) — compile-verified
//
#include <hip/hip_runtime.h>

#define N_USERS 200000
#define N_ITEMS 300000
#define NN      500000          // N_USERS + N_ITEMS
#define DD      64
#define EE      1200000
#define EPSF    0.1f
#define NORM_MIN 1e-12f

typedef float v2f __attribute__((ext_vector_type(2)));
typedef float v8f __attribute__((ext_vector_type(8)));
typedef float f2v __attribute__((ext_vector_type(2)));

// ---------------------------------------------------------------------------
// SpMM scatter: out[rows[e]] += vals[e] * ego[cols[e]]   (f32 atomics)
// 16 threads per edge, one float4 (16B) chunk each -> fully coalesced 256B
// row gathers. ego (~128MB) stays L2-resident (192MB L2) because all the
// streaming traffic elsewhere is marked non-temporal.
// ---------------------------------------------------------------------------
__global__ __launch_bounds__(256) void spmm_scatter(
    const int*   __restrict__ rows,
    const int*   __restrict__ cols,
    const float* __restrict__ vals,
    const float* __restrict__ ego,
    float*       __restrict__ out)
{
    size_t t = (size_t)blockIdx.x * blockDim.x + threadIdx.x;
    size_t e = t >> 4;
    if (e >= (size_t)EE) return;
    int q = (int)(t & 15);

    int   c = cols[e];
    int   r = rows[e];
    float v = vals[e];

    const float4* src = (const float4*)(ego + (size_t)c * DD) + q;
    float4 x = *src;

    float* dst = out + (size_t)r * DD + (size_t)q * 4;
    atomicAdd(dst + 0, v * x.x);
    atomicAdd(dst + 1, v * x.y);
    atomicAdd(dst + 2, v * x.z);
    atomicAdd(dst + 3, v * x.w);
}

// ---------------------------------------------------------------------------
// Perturb + accumulate. One wave per 16-row tile; 2 waves / 64-thread block;
// grid is exact (500000/16/2 = 15625 blocks) so EXEC is all-1s throughout.
//
// 1) Wave stages its 4KB noise tile into LDS with 8x
//    global_load_async_to_lds_b128 (TH=NT: stream, don't pollute L2),
//    then s_wait_asynccnt 0. Only this wave reads what it wrote.
// 2) Row-norm^2 via V_WMMA_F32_16X16X4_F32:
//    A (16x4, ISA layout): lane L owns row M=L&15; VGPR0 = local partial
//    sum-of-squares (K-slot 0 for lanes 0-15, slot 2 for lanes 16-31),
//    VGPR1 = 0. B = all-ones (layout-independent). D[i][*] = ||n_i||^2.
// 3) Update from LDS tile; tmp/acc accessed non-temporally.
// ---------------------------------------------------------------------------
__global__ __launch_bounds__(64) void perturb_acc(
    const float* __restrict__ se,    // SpMM result for this layer [NN, 64]
    const float* __restrict__ nz,    // noise_k                    [NN, 64]
    float*       __restrict__ ego,   // perturbed ego (next layer input)
    float*       __restrict__ acc,   // running accumulator (d_out)
    int mode)                        // 0: acc = e; 1: acc += e; 2: acc = (acc+e)/3
{
    __shared__ __align__(16) float sh[2][16 * DD];   // 4KB per wave

    int waveInBlk = (int)(threadIdx.x >> 5);
    int lane      = (int)(threadIdx.x & 31);
    int wave      = (int)blockIdx.x * 2 + waveInBlk;
    size_t r0     = (size_t)wave * 16;

    float* shw = sh[waveInBlk];

    // ---- stage 4KB noise tile into LDS via async copy (8 x 512B) ----------
    {
        unsigned long long g =
            (unsigned long long)(uintptr_t)((const char*)(nz + r0 * DD) + 16 * lane);
        unsigned int l =
            (unsigned int)(uintptr_t)shw + 16u * (unsigned)lane;
        asm volatile(
            "global_load_async_to_lds_b128 %0, %1, off offset:0    th:TH_LOAD_NT\n\t"
            "global_load_async_to_lds_b128 %0, %1, off offset:512  th:TH_LOAD_NT\n\t"
            "global_load_async_to_lds_b128 %0, %1, off offset:1024 th:TH_LOAD_NT\n\t"
            "global_load_async_to_lds_b128 %0, %1, off offset:1536 th:TH_LOAD_NT\n\t"
            "global_load_async_to_lds_b128 %0, %1, off offset:2048 th:TH_LOAD_NT\n\t"
            "global_load_async_to_lds_b128 %0, %1, off offset:2560 th:TH_LOAD_NT\n\t"
            "global_load_async_to_lds_b128 %0, %1, off offset:3072 th:TH_LOAD_NT\n\t"
            "global_load_async_to_lds_b128 %0, %1, off offset:3584 th:TH_LOAD_NT\n\t"
            "s_wait_asynccnt 0x0"
            :
            : "v"(l), "v"(g)
            : "memory");
    }

    // ---- phase 1: per-lane partial sum-of-squares, A-matrix layout --------
    int m  = lane & 15;
    int h0 = (lane < 16) ? 0 : 2;
    const float* srow = shw + m * DD;
    float s = 0.f;
#pragma unroll
    for (int c = 0; c < 16; ++c) {
        f2v p = *(const f2v*)(srow + 4 * c + h0);      // ds_load_b64
        s = fmaf(p.x, p.x, fmaf(p.y, p.y, s));
    }

    v2f a; a[0] = s;   a[1] = 0.f;
    v2f b; b[0] = 1.f; b[1] = 1.f;
    v8f cz = {};
    v8f dm = __builtin_amdgcn_wmma_f32_16x16x4_f32(
        /*neg_a=*/false, a, /*neg_b=*/false, b,
        /*c_mod=*/(short)0, cz, /*reuse_a=*/false, /*reuse_b=*/false);

    // per-row factor: eps / max(||n||, 1e-12)
    float inv[8];
#pragma unroll
    for (int v = 0; v < 8; ++v)
        inv[v] = EPSF / fmaxf(sqrtf(dm[v]), NORM_MIN);

    // ---- phase 2: ego = se + sign(se) * n * f ; accumulate (NT streams) ---
    const float third = 1.f / 3.f;
#pragma unroll
    for (int i = 0; i < 16; ++i) {
        // row i's factor is in VGPR (i&7) of lanes 0-15 (i<8) / 16-31 (i>=8)
        float fi = __shfl(inv[i & 7], (i < 8) ? 0 : 16, 32);

        size_t base = (r0 + (size_t)i) * DD + (size_t)(2 * lane);
        f2v nv = *(const f2v*)(shw + i * DD + 2 * lane);               // LDS
        f2v ev = __builtin_nontemporal_load((const f2v*)(se + base));  // NT

        float sx = (float)(ev.x > 0.f) - (float)(ev.x < 0.f);   // sign(0)=0
        float sy = (float)(ev.y > 0.f) - (float)(ev.y < 0.f);
        ev.x = fmaf(sx * nv.x, fi, ev.x);
        ev.y = fmaf(sy * nv.y, fi, ev.y);

        *(f2v*)(ego + base) = ev;       // RT: gathered by next layer's SpMM

        f2v av;
        if (mode == 0) {
            av = ev;
        } else {
            av = __builtin_nontemporal_load((const f2v*)(acc + base));
            av.x += ev.x; av.y += ev.y;
            if (mode == 2) { av.x *= third; av.y *= third; }
        }
        __builtin_nontemporal_store(av, (f2v*)(acc + base));
    }
}

// ---------------------------------------------------------------------------
// Host launcher. d_in order: user_emb, item_emb, vals, noise, rows, cols.
// d_ws layout: [0 .. NN*64)        = ego buffer   (128 MB)
//              [NN*64 .. 2*NN*64)  = spmm output  (128 MB)
// d_out doubles as the layer accumulator.
// ---------------------------------------------------------------------------
extern "C" void kernel_launch(void* const* d_in, const int* in_sizes, int n_in,
                              void* d_out, int out_size, void* d_ws, size_t ws_size,
                              hipStream_t stream) {
    const float* user_emb = (const float*)d_in[0];
    const float* item_emb = (const float*)d_in[1];
    const float* vals     = (const float*)d_in[2];
    const float* noise    = (const float*)d_in[3];
    const int*   rows     = (const int*)d_in[4];
    const int*   cols     = (const int*)d_in[5];

    float* acc = (float*)d_out;
    float* ego = (float*)d_ws;
    float* tmp = ego + (size_t)NN * DD;

    // ego = concat(user_emb, item_emb)
    hipMemcpyAsync(ego, user_emb,
                   (size_t)N_USERS * DD * sizeof(float),
                   hipMemcpyDeviceToDevice, stream);
    hipMemcpyAsync(ego + (size_t)N_USERS * DD, item_emb,
                   (size_t)N_ITEMS * DD * sizeof(float),
                   hipMemcpyDeviceToDevice, stream);

    dim3 sb(256);
    dim3 sg((unsigned)(((size_t)EE * 16 + 255) / 256));   // 75000 blocks

    dim3 pb(64);                                          // 2 waves / block
    dim3 pg((unsigned)(NN / 16 / 2));                     // 15625, exact

    for (int k = 0; k < 3; ++k) {
        hipMemsetAsync(tmp, 0, (size_t)NN * DD * sizeof(float), stream);
        spmm_scatter<<<sg, sb, 0, stream>>>(rows, cols, vals, ego, tmp);
        perturb_acc<<<pg, pb, 0, stream>>>(tmp, noise + (size_t)k * NN * DD,
                                           ego, acc, k);
    }
}